// GraphSAGEModel_51642686767944
// MI455X (gfx1250) — compile-verified
//
#include <hip/hip_runtime.h>
#include <hip/hip_bf16.h>
#include <math.h>

// ---------------------------------------------------------------------------
// GraphSAGE (3x SAGEConv with scatter-max aggregation) + log_softmax
//   agg = segment_max(x[src], dst);  agg[!finite] = 0
//   h   = agg @ Wl + bl + x @ Wr    (ReLU on layers 1,2)
//   out = log_softmax(h3, axis=1)
//
// Strategy for MI455X:
//  - scatter-max: order-preserving uint32 encoding of f32 so a single
//    hardware global_atomic_max_u32 does the reduction (works for negatives,
//    -inf is the natural identity).  Pure HBM-bandwidth bound.
//  - GEMMs: v_wmma_f32_16x16x4_f32, one 16x32 output tile per wave,
//    accumulating both agg@Wl and x@Wr into one bias-initialized accumulator.
//    A tiles staged in LDS (stride KPAD+1 -> conflict-free), weights
//    zero-padded in K (50 -> 52).
// ---------------------------------------------------------------------------

typedef __attribute__((ext_vector_type(2))) float v2f;
typedef __attribute__((ext_vector_type(8))) float v8f;

#define ENC_NEG_INF 0x007FFFFFu   // enc(-inf)

__device__ __forceinline__ unsigned enc_f32(float f) {
    unsigned u = __float_as_uint(f);
    return (u & 0x80000000u) ? ~u : (u | 0x80000000u);
}
__device__ __forceinline__ float dec_f32(unsigned k) {
    unsigned u = (k & 0x80000000u) ? (k & 0x7FFFFFFFu) : ~k;
    return __uint_as_float(u);
}

// ----------------------------- init agg ------------------------------------
__global__ __launch_bounds__(256) void fill_enc_neginf(unsigned* __restrict__ p, int n) {
    int i = blockIdx.x * blockDim.x + threadIdx.x;
    if (i < n) p[i] = ENC_NEG_INF;
}

// --------------------------- scatter-max -----------------------------------
// thread per (edge, channel): gather feat[src] channel, atomic-max into agg[dst]
__global__ __launch_bounds__(256) void scatter_max_kernel(
    const float* __restrict__ feat,          // [N, d]
    const long long* __restrict__ ei,        // [2, E] int64 (src row, dst row)
    unsigned* __restrict__ agg,              // [N, d] encoded keys
    int E, int d)
{
    int tid = blockIdx.x * blockDim.x + threadIdx.x;
    int total = E * d;
    if (tid >= total) return;
    int e = tid / d;
    int c = tid - e * d;
    long long s = ei[e];
    long long t = ei[(long long)E + e];
    float v = feat[s * (long long)d + c];
    atomicMax(&agg[t * (long long)d + c], enc_f32(v));
}

// ------------------------------ WMMA GEMM ----------------------------------
// out[N,32] = sanitize(dec(agg)) @ Wl + bl + xin @ Wr   (optional ReLU)
// block = 256 threads = 8 waves; wave w -> rows [blk*128 + 16w, +16), cols 0..31
template<int DIN, int KPAD, bool RELU>
__global__ __launch_bounds__(256) void sage_gemm_wmma(
    const unsigned* __restrict__ agg,   // [N, DIN] encoded
    const float*    __restrict__ xin,   // [N, DIN]
    const float*    __restrict__ Wl,    // [DIN, 32]
    const float*    __restrict__ Wr,    // [DIN, 32]
    const float*    __restrict__ bl,    // [32]
    float*          __restrict__ out,   // [N, 32]
    int N)
{
    constexpr int AS = KPAD + 1;        // LDS row stride, gcd(AS,64)==1
    __shared__ float sWl[KPAD][32];
    __shared__ float sWr[KPAD][32];
    __shared__ float sB[32];
    __shared__ float sAgg[8][16][AS];
    __shared__ float sX  [8][16][AS];

    const int tid  = threadIdx.x;
    const int wave = tid >> 5;
    const int lane = tid & 31;
    const int half = lane >> 4;         // 0: lanes 0-15, 1: lanes 16-31
    const int lm   = lane & 15;

    // --- stage weights (K zero-padded) + bias ---
    for (int i = tid; i < KPAD * 32; i += 256) {
        int k = i >> 5, n = i & 31;
        float wl = 0.f, wr = 0.f;
        if (k < DIN) { wl = Wl[k * 32 + n]; wr = Wr[k * 32 + n]; }
        sWl[k][n] = wl;
        sWr[k][n] = wr;
    }
    if (tid < 32) sB[tid] = bl[tid];

    // --- stage this wave's 16-row A tiles (agg decoded+sanitized, and x) ---
    const int rowBase = blockIdx.x * 128 + wave * 16;
    for (int i = lane; i < 16 * KPAD; i += 32) {
        int r = i / KPAD, c = i - r * KPAD;
        int row = rowBase + r;
        if (row >= N) row = N - 1;      // clamped; result discarded later
        float av = 0.f, xv = 0.f;
        if (c < DIN) {
            float a = dec_f32(agg[(long long)row * DIN + c]);
            av = isfinite(a) ? a : 0.f; // empty segments (-inf) -> 0
            xv = xin[(long long)row * DIN + c];
        }
        sAgg[wave][r][c] = av;
        sX  [wave][r][c] = xv;
    }
    __syncthreads();

    // --- accumulators: C/D layout row = v + 8*half, col = (n0) + lm ---
    v8f c0, c1;
    {
        float b0 = sB[lm], b1 = sB[16 + lm];
        #pragma unroll
        for (int v = 0; v < 8; ++v) { c0[v] = b0; c1[v] = b1; }
    }

    // --- K loop, 16x16x4 fp32 WMMA; 4 wmma per step (2 GEMMs x 2 col tiles)
    #pragma unroll
    for (int k = 0; k < KPAD; k += 4) {
        const int ka = k + 2 * half;    // A/B layout: VGPR0=K=ka, VGPR1=K=ka+1
        v2f aA, aX, bL0, bL1, bR0, bR1;
        aA.x  = sAgg[wave][lm][ka];     aA.y  = sAgg[wave][lm][ka + 1];
        aX.x  = sX  [wave][lm][ka];     aX.y  = sX  [wave][lm][ka + 1];
        bL0.x = sWl[ka][lm];            bL0.y = sWl[ka + 1][lm];
        bL1.x = sWl[ka][16 + lm];       bL1.y = sWl[ka + 1][16 + lm];
        bR0.x = sWr[ka][lm];            bR0.y = sWr[ka + 1][lm];
        bR1.x = sWr[ka][16 + lm];       bR1.y = sWr[ka + 1][16 + lm];

        c0 = __builtin_amdgcn_wmma_f32_16x16x4_f32(false, aA, false, bL0,
                                                   (short)0, c0, false, false);
        c1 = __builtin_amdgcn_wmma_f32_16x16x4_f32(false, aA, false, bL1,
                                                   (short)0, c1, false, false);
        c0 = __builtin_amdgcn_wmma_f32_16x16x4_f32(false, aX, false, bR0,
                                                   (short)0, c0, false, false);
        c1 = __builtin_amdgcn_wmma_f32_16x16x4_f32(false, aX, false, bR1,
                                                   (short)0, c1, false, false);
    }

    // --- store (wave-uniform guard; N is a multiple of 16) ---
    if (rowBase < N) {
        #pragma unroll
        for (int v = 0; v < 8; ++v) {
            int row = rowBase + v + 8 * half;
            float o0 = c0[v], o1 = c1[v];
            if (RELU) { o0 = fmaxf(o0, 0.f); o1 = fmaxf(o1, 0.f); }
            out[(long long)row * 32 + lm]      = o0;
            out[(long long)row * 32 + 16 + lm] = o1;
        }
    }
}

// --------------------------- log_softmax (C=32) ----------------------------
__global__ __launch_bounds__(256) void log_softmax32(float* __restrict__ out, int N) {
    int row = blockIdx.x * blockDim.x + threadIdx.x;
    if (row >= N) return;
    float4* p = (float4*)(out + (long long)row * 32);
    float4 v[8];
    float m = -INFINITY;
    #pragma unroll
    for (int i = 0; i < 8; ++i) {
        v[i] = p[i];
        m = fmaxf(m, fmaxf(fmaxf(v[i].x, v[i].y), fmaxf(v[i].z, v[i].w)));
    }
    float s = 0.f;
    #pragma unroll
    for (int i = 0; i < 8; ++i) {
        s += __expf(v[i].x - m) + __expf(v[i].y - m) +
             __expf(v[i].z - m) + __expf(v[i].w - m);
    }
    float lg = m + __logf(s);
    #pragma unroll
    for (int i = 0; i < 8; ++i) {
        v[i].x -= lg; v[i].y -= lg; v[i].z -= lg; v[i].w -= lg;
        p[i] = v[i];
    }
}

// ------------------------------- launch ------------------------------------
extern "C" void kernel_launch(void* const* d_in, const int* in_sizes, int n_in,
                              void* d_out, int out_size, void* d_ws, size_t ws_size,
                              hipStream_t stream) {
    const float*     x   = (const float*)d_in[0];      // [N, 50]
    const long long* ei  = (const long long*)d_in[1];  // [2, E] int64
    const float* Wl1 = (const float*)d_in[2];
    const float* bl1 = (const float*)d_in[3];
    const float* Wr1 = (const float*)d_in[4];
    const float* Wl2 = (const float*)d_in[5];
    const float* bl2 = (const float*)d_in[6];
    const float* Wr2 = (const float*)d_in[7];
    const float* Wl3 = (const float*)d_in[8];
    const float* bl3 = (const float*)d_in[9];
    const float* Wr3 = (const float*)d_in[10];

    const int C   = in_sizes[3];               // 32
    const int FIN = in_sizes[2] / C;           // 50
    const int N   = in_sizes[0] / FIN;         // 100000
    const int E   = in_sizes[1] / 2;           // 1600000

    // workspace layout
    unsigned* agg = (unsigned*)d_ws;                       // N * 52 keys (reused)
    float*    hA  = (float*)d_ws + (size_t)N * 52;         // N * 32
    float*    hB  = hA + (size_t)N * 32;                   // N * 32
    float*    o   = (float*)d_out;                         // N * 32

    const int T = 256;
    const int gemmBlocks = (N + 127) / 128;

    // ---------------- layer 1: din = 50 -> 32, ReLU ----------------
    {
        int n = N * FIN;
        fill_enc_neginf<<<(n + T - 1) / T, T, 0, stream>>>(agg, n);
        int w = E * FIN;
        scatter_max_kernel<<<(w + T - 1) / T, T, 0, stream>>>(x, ei, agg, E, FIN);
        sage_gemm_wmma<50, 52, true><<<gemmBlocks, T, 0, stream>>>(
            agg, x, Wl1, Wr1, bl1, hA, N);
    }
    // ---------------- layer 2: 32 -> 32, ReLU ----------------
    {
        int n = N * C;
        fill_enc_neginf<<<(n + T - 1) / T, T, 0, stream>>>(agg, n);
        int w = E * C;
        scatter_max_kernel<<<(w + T - 1) / T, T, 0, stream>>>(hA, ei, agg, E, C);
        sage_gemm_wmma<32, 32, true><<<gemmBlocks, T, 0, stream>>>(
            agg, hA, Wl2, Wr2, bl2, hB, N);
    }
    // ---------------- layer 3: 32 -> 32, no ReLU -> d_out ----------------
    {
        int n = N * C;
        fill_enc_neginf<<<(n + T - 1) / T, T, 0, stream>>>(agg, n);
        int w = E * C;
        scatter_max_kernel<<<(w + T - 1) / T, T, 0, stream>>>(hB, ei, agg, E, C);
        sage_gemm_wmma<32, 32, false><<<gemmBlocks, T, 0, stream>>>(
            agg, hB, Wl3, Wr3, bl3, o, N);
    }
    // ---------------- log_softmax over 32 channels, in place --------------
    log_softmax32<<<(N + T - 1) / T, T, 0, stream>>>(o, N);
}